// FastSoftmax_12283606468012
// MI455X (gfx1250) — compile-verified
//
#include <hip/hip_runtime.h>
#include <hip/hip_bf16.h>

typedef float f32x2 __attribute__((ext_vector_type(2)));
typedef int   v2i   __attribute__((ext_vector_type(2)));

#define WAVES_PER_BLOCK 8
#define BLOCKS 4096
#define ROW_PAIRS 256          // 2048 B / 8 B : max row (506 f32 = 2024 B) padded
#define COPIES_PER_ROW 8       // 8 x b64 x 32 lanes = 2048 B per row
#define NEG_HUGE (-3.402823466e38f)

// ---- CDNA5 async global->LDS copy + ASYNCcnt wait --------------------------
// Probe result (round 1 diagnostic): builtin param0 is a non-const AS(1)
// pointer to a 2 x i32 vector; match that exactly.
#if __has_builtin(__builtin_amdgcn_global_load_async_to_lds_b64)
#define ASYNC_COPY_B64(gp, lp)                                                \
  __builtin_amdgcn_global_load_async_to_lds_b64(                              \
      (__attribute__((address_space(1))) v2i*)(void*)(gp),                    \
      (__attribute__((address_space(3))) v2i*)(void*)(lp), 0, 0)
#else
#define ASYNC_COPY_B64(gp, lp)                                                \
  asm volatile("global_load_async_to_lds_b64 %0, %1, off"                     \
               :: "v"((unsigned)(unsigned long long)                          \
                       (__attribute__((address_space(3))) char*)(void*)(lp)), \
                  "v"((unsigned long long)(const void*)(gp))                  \
               : "memory")
#endif

#if __has_builtin(__builtin_amdgcn_s_wait_asynccnt)
#define WAIT_ASYNC(n)                                                         \
  do { __builtin_amdgcn_s_wait_asynccnt(n); asm volatile("" ::: "memory"); } while (0)
#else
#define WAIT_ASYNC(n) asm volatile("s_wait_asynccnt %0" :: "i"(n) : "memory")
#endif
// ----------------------------------------------------------------------------

// ws layout: [0..batch] = row prefix (rows before sample b, rows = heads*s_b)
//            [batch+1 .. 2*batch+1] = element prefix (heads * s_b^2)
__global__ void build_prefix_kernel(const int* __restrict__ seqlen,
                                    const int* __restrict__ headsArr,
                                    unsigned* __restrict__ ws, int batch) {
  if (threadIdx.x == 0 && blockIdx.x == 0) {
    unsigned h = (unsigned)headsArr[0];
    unsigned rp = 0, ep = 0;
    for (int b = 0; b < batch; ++b) {
      ws[b] = rp;
      ws[batch + 1 + b] = ep;
      unsigned s = (unsigned)seqlen[b];
      rp += h * s;
      ep += h * s * s;
    }
    ws[batch] = rp;                    // total rows R
    ws[batch + 1 + batch] = ep;        // total elements T
  }
}

__global__ __launch_bounds__(256)
void ragged_softmax_kernel(const float* __restrict__ x,
                           const int* __restrict__ seqlen,
                           const unsigned* __restrict__ ws,
                           float* __restrict__ out, int batch) {
  __shared__ f32x2 smem[WAVES_PER_BLOCK * 2 * ROW_PAIRS];   // 32 KB / WG

  const int lane  = threadIdx.x & 31;
  const int wslot = threadIdx.x >> 5;

  const unsigned* rowPref  = ws;
  const unsigned* elemPref = ws + batch + 1;
  const unsigned R       = rowPref[batch];
  const unsigned maxPair = (elemPref[batch] >> 1) - 1u;     // clamp for buffer end
  const unsigned stride  = gridDim.x * WAVES_PER_BLOCK;

  const f32x2* x2 = (const f32x2*)x;
  f32x2* buf0 = smem + (wslot * 2) * ROW_PAIRS;
  f32x2* buf1 = buf0 + ROW_PAIRS;

  // uniform row -> (length, element base) via binary search over row prefix
  auto lookup = [&](unsigned row, unsigned& s, unsigned& base) {
    unsigned ru = (unsigned)__builtin_amdgcn_readfirstlane((int)row);
    unsigned lo = 0, hi = (unsigned)batch;
    while (hi - lo > 1u) {
      unsigned mid = (lo + hi) >> 1;
      if (rowPref[mid] <= ru) lo = mid; else hi = mid;
    }
    s    = (unsigned)seqlen[lo];
    base = elemPref[lo] + (ru - rowPref[lo]) * s;
  };

  // stream one row (2 KB, fixed 8 async b64 instructions) into an LDS buffer
  auto issue = [&](unsigned base, f32x2* lbuf) {
    unsigned bp = base >> 1;           // row base is always even (seqlens even)
#pragma unroll
    for (int c = 0; c < COPIES_PER_ROW; ++c) {
      unsigned p = bp + (unsigned)(c * 32 + lane);
      p = p > maxPair ? maxPair : p;   // never read past the packed buffer
      ASYNC_COPY_B64(x2 + p, lbuf + c * 32 + lane);
    }
  };

  unsigned r = blockIdx.x * WAVES_PER_BLOCK + wslot;
  unsigned curS = 0, curBase = 0, nxtS = 0, nxtBase = 0;
  int cur = 0;

  if (r < R) { lookup(r, curS, curBase); issue(curBase, buf0); }

  while (r < R) {
    unsigned rn = r + stride;
    f32x2* cbuf = cur ? buf1 : buf0;
    f32x2* nbuf = cur ? buf0 : buf1;

    if (rn < R) {
      lookup(rn, nxtS, nxtBase);
      issue(nxtBase, nbuf);            // prefetch next row, overlap with compute
      WAIT_ASYNC(COPIES_PER_ROW);      // in-order retire => current row landed
    } else {
      WAIT_ASYNC(0);
    }

    // ---- softmax of current row, read once from LDS ----
    {
      const unsigned s = curS;
      f32x2 v[8];
      float m = NEG_HUGE;
#pragma unroll
      for (int c = 0; c < 8; ++c) {
        unsigned e0 = (unsigned)(c * 64) + (unsigned)(lane * 2);
        f32x2 val;
        if (e0 < s) val = cbuf[c * 32 + lane];            // ds_load_b64
        else { val.x = NEG_HUGE; val.y = NEG_HUGE; }
        v[c] = val;
        m = fmaxf(m, fmaxf(val.x, val.y));
      }
#pragma unroll
      for (int o = 16; o > 0; o >>= 1) m = fmaxf(m, __shfl_xor(m, o, 32));

      float sum = 0.f;
#pragma unroll
      for (int c = 0; c < 8; ++c) {
        if ((unsigned)(c * 64) < s) {                     // uniform chunk guard
          float ex = __expf(v[c].x - m);
          float ey = __expf(v[c].y - m);
          v[c].x = ex; v[c].y = ey;
          sum += ex + ey;                                 // masked lanes add 0
        }
      }
#pragma unroll
      for (int o = 16; o > 0; o >>= 1) sum += __shfl_xor(sum, o, 32);
      float inv = 1.0f / sum;

#pragma unroll
      for (int c = 0; c < 8; ++c) {
        unsigned e0 = (unsigned)(c * 64) + (unsigned)(lane * 2);
        if (e0 < s) {
          f32x2 o2 = v[c];
          o2.x *= inv; o2.y *= inv;
          __builtin_nontemporal_store(o2, (f32x2*)(out + curBase + e0));
        }
      }
    }

    curS = nxtS; curBase = nxtBase;
    r = rn; cur ^= 1;
  }
}

extern "C" void kernel_launch(void* const* d_in, const int* in_sizes, int n_in,
                              void* d_out, int out_size, void* d_ws, size_t ws_size,
                              hipStream_t stream) {
  // setup_inputs order: input, seqlen, row_ids, num_rows, heads, batch
  const float* x      = (const float*)d_in[0];
  const int*   seqlen = (const int*)d_in[1];
  const int*   heads  = (const int*)d_in[4];
  int batch = in_sizes[1];                    // seqlen has `batch` entries
  unsigned* ws = (unsigned*)d_ws;

  hipLaunchKernelGGL(build_prefix_kernel, dim3(1), dim3(32), 0, stream,
                     seqlen, heads, ws, batch);
  hipLaunchKernelGGL(ragged_softmax_kernel, dim3(BLOCKS),
                     dim3(WAVES_PER_BLOCK * 32), 0, stream,
                     x, seqlen, (const unsigned*)ws, (float*)d_out, batch);
}